// WeightedAttention_35081292874263
// MI455X (gfx1250) — compile-verified
//
#include <hip/hip_runtime.h>

typedef __attribute__((ext_vector_type(2))) float v2f;
typedef __attribute__((ext_vector_type(8))) float v8f;

#define B_ 32
#define S_ 2048
#define D_ 512
#define H_ 32
#define ROWS_ (B_ * S_)      // 65536 token rows
#define SCHUNKS_ 8

__device__ __forceinline__ float sigmoidf_(float x) {
    return 1.0f / (1.0f + __expf(-x));
}

// ---------------------------------------------------------------------------
// Kernel 1: per-token scores.  Each wave computes a 16-row tile of
// sigmoid((x @ P)/s) with V_WMMA_F32_16X16X4_F32 (two N-tiles for H=32),
// then the 32x32 hidden layer + evaluator dot in VALU via LDS.
// ---------------------------------------------------------------------------
__global__ __launch_bounds__(128) void score_kernel(
    const float* __restrict__ inp,          // [B,S,D]
    const unsigned char* __restrict__ mask, // [B,S] bool
    const float* __restrict__ proj,         // [D,H]
    const float* __restrict__ hid,          // [1,H,H]
    const float* __restrict__ evalv,        // [H,1]
    float* __restrict__ scores)             // [B*S]
{
    const float inv_scale = 0.17677669529663687f; // 1/sqrt(32)
    __shared__ float w0t[4][16][33];              // per-wave 16x32 tile (+pad)

    const int tid  = threadIdx.x;
    const int wave = tid >> 5;
    const int lane = tid & 31;
    const int rowbase = (blockIdx.x * 4 + wave) * 16;

    const int mrow  = lane & 15;       // M index within tile (A frag / C cols)
    const int khalf = (lane >> 4) * 2; // K sub-offset per ISA A/B layout
    const int r = rowbase + mrow;

    const float* xrow = inp + (size_t)r * D_;
    const float  mf   = mask[r] ? 1.0f : 0.0f;   // x = inp * m

    v8f c0 = {}; v8f c1 = {};
    for (int k = 0; k < D_; k += 4) {
        // A frag: lanes 0-15 hold {K=k,k+1}, lanes 16-31 hold {K=k+2,k+3}
        v2f a;
        const float* ap = xrow + k + khalf;
        a.x = ap[0] * mf;
        a.y = ap[1] * mf;
        // B frags (projector rows k..k+3), N-tile 0 = cols 0-15, tile 1 = 16-31
        const float* bp = proj + (size_t)(k + khalf) * H_;
        v2f b0, b1;
        b0.x = bp[mrow];        b0.y = bp[H_ + mrow];
        b1.x = bp[16 + mrow];   b1.y = bp[H_ + 16 + mrow];
        c0 = __builtin_amdgcn_wmma_f32_16x16x4_f32(false, a, false, b0,
                                                   (short)0, c0, false, false);
        c1 = __builtin_amdgcn_wmma_f32_16x16x4_f32(false, a, false, b1,
                                                   (short)0, c1, false, false);
    }

    // C layout: VGPR i -> row = i + 8*(lane>=16), col = lane&15
    const int rhalf = (lane >> 4) * 8;
#pragma unroll
    for (int i = 0; i < 8; ++i) {
        w0t[wave][rhalf + i][mrow]      = sigmoidf_(c0[i] * inv_scale);
        w0t[wave][rhalf + i][16 + mrow] = sigmoidf_(c1[i] * inv_scale);
    }
    __syncthreads();

    // Hidden 32x32 + evaluator: lane j computes w1[r][j], butterfly-reduce.
    const float ej = evalv[lane];
    for (int rr = 0; rr < 16; ++rr) {
        float t = 0.0f;
#pragma unroll
        for (int h = 0; h < H_; ++h)
            t = fmaf(w0t[wave][rr][h], hid[h * H_ + lane], t);
        float contrib = sigmoidf_(t * inv_scale) * ej;
#pragma unroll
        for (int off = 16; off >= 1; off >>= 1)
            contrib += __shfl_xor(contrib, off, 32);
        if (lane == 0)
            scores[rowbase + rr] = sigmoidf_(contrib * inv_scale);
    }
}

// ---------------------------------------------------------------------------
// Kernel 2: per-batch softmax over S with mask + renorm.
// final weight = m*e(s-max) / (sum(m*e) + 1e-12 * sum(e))
// ---------------------------------------------------------------------------
__global__ __launch_bounds__(256) void softmax_kernel(
    const float* __restrict__ scores,
    const unsigned char* __restrict__ mask,
    float* __restrict__ weights)
{
    __shared__ float red[256];
    const int b = blockIdx.x, tid = threadIdx.x;
    const float* sc = scores + (size_t)b * S_;
    const unsigned char* mk = mask + (size_t)b * S_;

    float lmax = -3.0e38f;
    for (int s = tid; s < S_; s += 256) lmax = fmaxf(lmax, sc[s]);
    red[tid] = lmax; __syncthreads();
    for (int st = 128; st >= 1; st >>= 1) {
        if (tid < st) red[tid] = fmaxf(red[tid], red[tid + st]);
        __syncthreads();
    }
    const float mx = red[0]; __syncthreads();

    float z = 0.0f, mz = 0.0f;
    for (int s = tid; s < S_; s += 256) {
        float e = __expf(sc[s] - mx);
        z += e;
        mz += mk[s] ? e : 0.0f;
    }
    red[tid] = z; __syncthreads();
    for (int st = 128; st >= 1; st >>= 1) {
        if (tid < st) red[tid] += red[tid + st];
        __syncthreads();
    }
    const float Z = red[0]; __syncthreads();
    red[tid] = mz; __syncthreads();
    for (int st = 128; st >= 1; st >>= 1) {
        if (tid < st) red[tid] += red[tid + st];
        __syncthreads();
    }
    const float MZ = red[0]; __syncthreads();

    const float inv_den = 1.0f / (MZ + 1e-12f * Z);
    float* w = weights + (size_t)b * S_;
    for (int s = tid; s < S_; s += 256) {
        float e = __expf(sc[s] - mx);
        w[s] = mk[s] ? e * inv_den : 0.0f;
    }
}

// ---------------------------------------------------------------------------
// Kernel 3: partial weighted sums over S chunks (deterministic, no atomics).
// weight already carries the mask, so multiply raw inp.
// ---------------------------------------------------------------------------
__global__ __launch_bounds__(256) void wsum_kernel(
    const float* __restrict__ inp,
    const float* __restrict__ weights,
    float* __restrict__ partials)           // [B, SCHUNKS, D]
{
    const int b  = blockIdx.x;
    const int dc = blockIdx.y;              // 0..1 (256 cols each)
    const int sc = blockIdx.z;              // 0..SCHUNKS_-1
    const int d  = dc * 256 + threadIdx.x;

    const float* base = inp + ((size_t)b * S_) * D_ + d;
    const float* w    = weights + (size_t)b * S_;
    const int s0 = sc * (S_ / SCHUNKS_);
    const int s1 = s0 + (S_ / SCHUNKS_);

    float acc = 0.0f;
    for (int s = s0; s < s1; ++s)
        acc = fmaf(w[s], base[(size_t)s * D_], acc);
    partials[(((size_t)b * SCHUNKS_) + sc) * D_ + d] = acc;
}

// ---------------------------------------------------------------------------
// Kernel 4: reduce partials -> out[B,D]
// ---------------------------------------------------------------------------
__global__ __launch_bounds__(256) void reduce_kernel(
    const float* __restrict__ partials,
    float* __restrict__ out)
{
    const int i = blockIdx.x * 256 + threadIdx.x;  // b*D + d
    const int b = i / D_, d = i % D_;
    float acc = 0.0f;
#pragma unroll
    for (int sc = 0; sc < SCHUNKS_; ++sc)
        acc += partials[(((size_t)b * SCHUNKS_) + sc) * D_ + d];
    out[i] = acc;
}

extern "C" void kernel_launch(void* const* d_in, const int* in_sizes, int n_in,
                              void* d_out, int out_size, void* d_ws, size_t ws_size,
                              hipStream_t stream) {
    (void)in_sizes; (void)n_in; (void)out_size; (void)ws_size;
    const float*         inp   = (const float*)d_in[0];
    const unsigned char* mask  = (const unsigned char*)d_in[1];
    const float*         proj  = (const float*)d_in[2];
    const float*         hid   = (const float*)d_in[3];
    const float*         evalv = (const float*)d_in[4];

    float* scores   = (float*)d_ws;           // [B*S]
    float* weights  = scores + ROWS_;         // [B*S]
    float* partials = weights + ROWS_;        // [B*SCHUNKS_*D]

    score_kernel<<<ROWS_ / (16 * 4), 128, 0, stream>>>(inp, mask, proj, hid,
                                                       evalv, scores);
    softmax_kernel<<<B_, 256, 0, stream>>>(scores, mask, weights);
    dim3 g3(B_, D_ / 256, SCHUNKS_);
    wsum_kernel<<<g3, 256, 0, stream>>>(inp, weights, partials);
    reduce_kernel<<<(B_ * D_) / 256, 256, 0, stream>>>(partials, (float*)d_out);
}